// Drug_Cell_In_74715251081389
// MI455X (gfx1250) — compile-verified
//
#include <hip/hip_runtime.h>

// ---------------------------------------------------------------------------
// out[b,n] = softmax_n( cc_b . dhat[b,n] ) * ( cn_b . dhat[b,n] ),
//   dhat = d/||d||, cn = cell/||cell||, u = cn @ q, cc = k @ u.
// Algebraic rewrite kills the [B*N,F]x[F,F] GEMM (34.4 GFLOP -> 0.6 GFLOP):
// HBM-bound on the 256MB drug tensor (~11us @ 23.3 TB/s).
// G=4 batches per block amortizes q/k L2 traffic 4x (1GB -> 256MB).
// Dual dot products per drug row via V_WMMA_F32_16X16X4_F32 (full fp32),
// 16 rows per wave32; row norms folded in as a post-scale on the C rows.
// ---------------------------------------------------------------------------

typedef __attribute__((ext_vector_type(2))) float v2f;
typedef __attribute__((ext_vector_type(4))) float v4f;
typedef __attribute__((ext_vector_type(8))) float v8f;

namespace {
constexpr int kN = 128;   // drug rows per batch
constexpr int kF = 256;   // feature dim
constexpr int kG = 4;     // batches per block
}

__global__ __launch_bounds__(256)
void drugcell_attn_kernel(const float* __restrict__ drug,
                          const float* __restrict__ cell,
                          const float* __restrict__ kmat,
                          const float* __restrict__ qmat,
                          float* __restrict__ out)
{
    __shared__ __align__(16) float s_cn4[kF * kG];          // interleaved [i][g]
    __shared__ __align__(16) float s_u4[kF * kG];           // interleaved [f][g]; also sq scratch
    __shared__ __align__(16) float s_ab[(2 * kG + 1) * kF]; // planar A panes: cc[g] | cn[g] | zeros
    __shared__ float s_score[kN];
    __shared__ float s_sim[kN];
    __shared__ float s_exp[kN];

    const int t  = threadIdx.x;        // 0..255
    const int b0 = blockIdx.x * kG;    // first batch of this block

    // ---- 1) cell rows: squares -> norms (4 batches at once) -------------
    float cv[kG];
#pragma unroll
    for (int g = 0; g < kG; ++g)
        cv[g] = cell[(size_t)(b0 + g) * kF + t];

    {
        v4f sq;
#pragma unroll
        for (int g = 0; g < kG; ++g) sq[g] = cv[g] * cv[g];
        *(v4f*)(&s_u4[t * kG]) = sq;                        // ds_store_b128
    }
    __syncthreads();

    v4f ss = {};
#pragma unroll 8
    for (int i = 0; i < kF; ++i)
        ss += *(const v4f*)(&s_u4[i * kG]);                 // ds_load_b128 broadcast

    v4f cnv;
#pragma unroll
    for (int g = 0; g < kG; ++g)
        cnv[g] = cv[g] * (1.0f / fmaxf(sqrtf(ss[g]), 1e-12f));
    *(v4f*)(&s_cn4[t * kG]) = cnv;
    __syncthreads();   // cn visible; all square reads done -> s_u4 reusable

    // ---- 2) u[g] = cn[g] @ q   (coalesced q stream, 4 accumulators) -----
    {
        v4f acc = {};
#pragma unroll 2
        for (int i = 0; i < kF; ++i) {
            const float qv = qmat[(size_t)i * kF + t];
            const v4f cn = *(const v4f*)(&s_cn4[i * kG]);
#pragma unroll
            for (int g = 0; g < kG; ++g) acc[g] = fmaf(cn[g], qv, acc[g]);
        }
        *(v4f*)(&s_u4[t * kG]) = acc;
    }
    __syncthreads();

    // ---- 3) cc[g] = k @ u[g]; lay out planar A panes --------------------
    {
        const float* krow = kmat + (size_t)t * kF;
        v4f acc = {};
#pragma unroll 2
        for (int f = 0; f < kF; ++f) {
            const float kv = krow[f];
            const v4f uv = *(const v4f*)(&s_u4[f * kG]);
#pragma unroll
            for (int g = 0; g < kG; ++g) acc[g] = fmaf(kv, uv[g], acc[g]);
        }
#pragma unroll
        for (int g = 0; g < kG; ++g) {
            s_ab[g * kF + t]        = acc[g];   // A row 0 pane: cc[g]
            s_ab[(kG + g) * kF + t] = cnv[g];   // A row 1 pane: cn[g]
        }
        s_ab[2 * kG * kF + t] = 0.0f;           // zero pane for A rows >= 2
    }
    __syncthreads();

    // ---- 4) per-batch drug tiles: WMMA + softmax ------------------------
    // B operand (4x16 KxN): lane n in [0,16) holds d[row n][k0..k0+1],
    // lane n+16 holds d[row n][k0+2..k0+3]. A (16x4): row0=cc, row1=cn, rest 0.
    const int lane = t & 31;
    const int wave = t >> 5;            // 0..7
    const int r    = lane & 15;         // tile row / A row / B column
    const int half = lane >> 4;         // K-pair selector
    const int row  = wave * 16 + r;     // 0..127

    for (int g = 0; g < kG; ++g) {
        const int arow = (r == 0) ? g : ((r == 1) ? (kG + g) : 2 * kG);
        const v2f* __restrict__ apane =
            (const v2f*)(&s_ab[arow * kF + half * 2]);
        const float* drow = drug + ((size_t)(b0 + g) * kN + row) * kF;

        // warm WGP$/L2: each lane prefetches 8 cachelines of its row (16KB/wave)
#pragma unroll
        for (int j = 0; j < 8; ++j)
            __builtin_prefetch(drow + (half * 8 + j) * 16, 0, 1);

        const float* dptr = drow + half * 2;
        v8f C = {};
        float ssq = 0.0f;
#pragma unroll 8
        for (int k0 = 0; k0 < kF; k0 += 4) {
            const v2f bv = *(const v2f*)(dptr + k0);        // global_load_b64
            ssq = fmaf(bv.x, bv.x, ssq);
            ssq = fmaf(bv.y, bv.y, ssq);
            const v2f av = apane[k0 >> 1];                  // ds_load_b64
            C = __builtin_amdgcn_wmma_f32_16x16x4_f32(
                    false, av, false, bv, (short)0, C, false, false);
        }

        // combine half-wave sum-of-squares partials -> row inverse norm
        const float ssq_o = __shfl_xor(ssq, 16, 32);
        const float dinv = 1.0f / fmaxf(sqrtf(ssq + ssq_o), 1e-12f);

        if (lane < 16) {
            // C: VGPR j, lanes 0..15 -> C[M=j][N=lane]; M=0 scores, M=1 sim
            s_score[row] = C[0] * dinv;
            s_sim[row]   = C[1] * dinv;
        }
        __syncthreads();

        float m = -3.402823466e+38f;
#pragma unroll 8
        for (int j = 0; j < kN; ++j) m = fmaxf(m, s_score[j]);

        float e = 0.0f, sv = 0.0f;
        if (t < kN) {
            e = __expf(s_score[t] - m);
            s_exp[t] = e;
            sv = s_sim[t];
        }
        __syncthreads();

        float sum = 0.0f;
#pragma unroll 8
        for (int j = 0; j < kN; ++j) sum += s_exp[j];

        if (t < kN)
            out[(size_t)(b0 + g) * kN + t] = (e / sum) * sv;
        __syncthreads();   // s_score/s_exp reused by next g
    }
}

extern "C" void kernel_launch(void* const* d_in, const int* in_sizes, int n_in,
                              void* d_out, int out_size, void* d_ws, size_t ws_size,
                              hipStream_t stream) {
    const float* drug = (const float*)d_in[0];   // [B, N, F]
    const float* cell = (const float*)d_in[1];   // [B, F]
    const float* kmat = (const float*)d_in[2];   // [F, F]
    const float* qmat = (const float*)d_in[3];   // [F, F]
    float* out = (float*)d_out;                  // [B, N]

    const int B = in_sizes[1] / kF;              // 2048
    drugcell_attn_kernel<<<dim3(B / kG), dim3(256), 0, stream>>>(
        drug, cell, kmat, qmat, out);
}